// GLASS_27685359190612
// MI455X (gfx1250) — compile-verified
//
#include <hip/hip_runtime.h>
#include <hip/hip_bf16.h>
#include <math.h>

#define N_NODES 50000
#define N_EDGES 800000
#define FEATD   128
#define HIDD    256
#define NSUB    512
#define SUBSZ   64
#define MBAT    (NSUB * SUBSZ)
#define K0PAD   132

typedef __attribute__((ext_vector_type(2))) float v2f;
typedef __attribute__((ext_vector_type(8))) float v8f;

// b128 payload type for the async global->LDS builtin (GCC vector of 4 ints,
// as revealed by the clang diagnostic), plus AS1/AS3 pointer flavors.
typedef int v4i_gcc __attribute__((vector_size(16)));
typedef __attribute__((address_space(1))) v4i_gcc* gas_v4i_ptr;
typedef __attribute__((address_space(3))) v4i_gcc* las_v4i_ptr;

// Probe-safe detection of the gfx1250 async global->LDS copy path.
#if defined(__HIP_DEVICE_COMPILE__) && \
    __has_builtin(__builtin_amdgcn_global_load_async_to_lds_b128) && \
    __has_builtin(__builtin_amdgcn_s_wait_asynccnt)
#define GLASS_ASYNC_LDS 1
#else
#define GLASS_ASYNC_LDS 0
#endif

// ---------------------------------------------------------------------------
// h0 = concat(x, batch_indicator), stored with stride 256 (zero-padded).
// ---------------------------------------------------------------------------
__global__ void build_h0_kernel(const float* __restrict__ x, float* __restrict__ h) {
  int idx = blockIdx.x * blockDim.x + threadIdx.x;
  if (idx >= N_NODES * HIDD) return;
  int row = idx >> 8;
  int col = idx & 255;
  h[idx] = (col < FEATD) ? x[row * FEATD + col] : 0.0f;
}

__global__ void set_batch_feat_kernel(const int* __restrict__ batches,
                                      float* __restrict__ h) {
  int m = blockIdx.x * blockDim.x + threadIdx.x;
  if (m >= MBAT) return;
  h[batches[m] * HIDD + FEATD] = 1.0f;  // duplicates all write 1.0 -> fine
}

__global__ void zero_kernel(float* __restrict__ p, int n) {
  int i = blockIdx.x * blockDim.x + threadIdx.x;
  if (i < n) p[i] = 0.0f;
}

// Pad W0 [129,256] -> Wp [132,256] with zero rows so the GEMM loop is unguarded.
__global__ void pad_w0_kernel(const float* __restrict__ W0, float* __restrict__ Wp) {
  int idx = blockIdx.x * blockDim.x + threadIdx.x;
  if (idx >= K0PAD * HIDD) return;
  int k = idx >> 8;
  Wp[idx] = (k < 129) ? W0[idx] : 0.0f;
}

// ---------------------------------------------------------------------------
// Edge scatter: msg[dst] += w * h[src].  One thread per (edge, float4 group).
// h (~51MB) is L2-resident (192MB L2) -> gather + f32 atomics run at L2 speed.
// ---------------------------------------------------------------------------
__global__ void scatter_edges_kernel(const int* __restrict__ src,
                                     const int* __restrict__ dst,
                                     const float* __restrict__ w,
                                     const float* __restrict__ h,
                                     float* __restrict__ msg,
                                     int ngroups, int total) {
  int tid = blockIdx.x * blockDim.x + threadIdx.x;
  if (tid >= total) return;
  int e = tid / ngroups;
  int g = tid - e * ngroups;
  int s = src[e];
  int d = dst[e];
  float wt = w[e];
  const float4 v = *(const float4*)(h + s * HIDD + g * 4);
  float* mp = msg + d * HIDD + g * 4;
  atomicAdd(mp + 0, wt * v.x);
  atomicAdd(mp + 1, wt * v.y);
  atomicAdd(mp + 2, wt * v.z);
  atomicAdd(mp + 3, wt * v.w);
}

// U = (1+eps[layer]) * h + msg   (written into msg buffer)
__global__ void combine_kernel(const float* __restrict__ h, float* __restrict__ msg,
                               const float* __restrict__ eps, int layer, int n) {
  int i = blockIdx.x * blockDim.x + threadIdx.x;
  if (i >= n) return;
  float e = 1.0f + eps[layer];
  msg[i] = e * h[i] + msg[i];
}

// ---------------------------------------------------------------------------
// out = relu(A @ W + bias), output N-dim fixed at 256, V_WMMA_F32_16X16X4_F32.
// 8 waves per block; all 8 share the same 16-row A-tile (tiles blockIdx*8..+7
// never cross a 16-tile row boundary), so the tile is staged once in LDS,
// via async global->LDS copy when the toolchain exposes it.
// ---------------------------------------------------------------------------
template <int K>
__global__ __launch_bounds__(256) void wmma_gemm_relu_kernel(
    const float* __restrict__ A, int lda,
    const float* __restrict__ W,
    const float* __restrict__ bias,
    float* __restrict__ out, int ldo) {
  __shared__ float As[16 * K];
  const int tid   = threadIdx.x;
  const int lane  = tid & 31;
  const int wave  = tid >> 5;
  const int tile  = blockIdx.x * 8 + wave;
  const int nt    = tile & 15;
  const int mt    = tile >> 4;            // identical for all 8 waves in block
  const int l15   = lane & 15;
  const int khalf = (lane >> 4) << 1;     // 0 or 2

  // Cooperative staging of the 16 x K A-tile (float4 granularity).
  const int nq = 4 * K;                   // total float4 quads (16 rows * K/4)
  for (int i = tid; i < nq; i += 256) {
    const int row = i / (K / 4);
    const int kq  = i - row * (K / 4);
    const float* gp = A + (mt * 16 + row) * lda + kq * 4;
    float* lp = &As[row * K + kq * 4];
#if GLASS_ASYNC_LDS
    __builtin_amdgcn_global_load_async_to_lds_b128(
        (gas_v4i_ptr)(uintptr_t)gp,
        (las_v4i_ptr)(uint32_t)(uintptr_t)lp,
        /*offset=*/0, /*cpol=*/0);
#else
    *(float4*)lp = *(const float4*)gp;
#endif
  }
#if GLASS_ASYNC_LDS
  __builtin_amdgcn_s_wait_asynccnt(0);
#endif
  __syncthreads();

  const float* arow = &As[l15 * K];       // LDS reads -> ds_load
  const int col = nt * 16 + l15;
  const float* wcol = W + col;

  v8f acc = {};
#pragma unroll 4
  for (int ks = 0; ks < K / 4; ++ks) {
    const int k = ks * 4 + khalf;
    v2f a, b;
    a[0] = arow[k];
    a[1] = arow[k + 1];
    b[0] = wcol[k * HIDD];
    b[1] = wcol[(k + 1) * HIDD];
    acc = __builtin_amdgcn_wmma_f32_16x16x4_f32(
        /*neg_a=*/false, a, /*neg_b=*/false, b,
        /*c_mod=*/(short)0, acc, /*reuse_a=*/false, /*reuse_b=*/false);
  }

  const int rbase = mt * 16 + ((lane >> 4) << 3);  // lanes 16-31 hold M+8 rows
  const float bb = bias[col];
#pragma unroll
  for (int v = 0; v < 8; ++v) {
    out[(rbase + v) * ldo + col] = fmaxf(acc[v] + bb, 0.0f);
  }
}

// ---------------------------------------------------------------------------
// MultiPool: segments are 64 consecutive gathered rows; one block per subgraph,
// one thread per hidden channel. z row = [sum | mean | min | max] (4*HID).
// ---------------------------------------------------------------------------
__global__ void multipool_kernel(const float* __restrict__ h,
                                 const int* __restrict__ batches,
                                 float* __restrict__ z) {
  int s = blockIdx.x;
  int j = threadIdx.x;
  float sum = 0.0f;
  float mn = 3.402823466e+38f;
  float mx = -3.402823466e+38f;
  const int* bp = batches + s * SUBSZ;
  for (int i = 0; i < SUBSZ; ++i) {
    float v = h[bp[i] * HIDD + j];
    sum += v;
    mn = fminf(mn, v);
    mx = fmaxf(mx, v);
  }
  float* zr = z + s * 4 * HIDD;
  zr[j]            = sum;
  zr[HIDD + j]     = sum * (1.0f / SUBSZ);   // count is exactly 64 per segment
  zr[2 * HIDD + j] = mn;
  zr[3 * HIDD + j] = mx;
}

// ---------------------------------------------------------------------------
// preds = r1 @ Wr2 + br2 (256->1), BCE-with-logits, mean over 512 graphs.
// ---------------------------------------------------------------------------
__global__ void loss_kernel(const float* __restrict__ r1,
                            const float* __restrict__ Wr2,
                            const float* __restrict__ br2,
                            const float* __restrict__ labels,
                            float* __restrict__ out) {
  __shared__ float red[NSUB];
  int s = threadIdx.x;
  float p = br2[0];
  for (int k = 0; k < HIDD; ++k) p += r1[s * HIDD + k] * Wr2[k];
  float y = labels[s];
  float term = fmaxf(p, 0.0f) - p * y + log1pf(expf(-fabsf(p)));
  red[s] = term;
  __syncthreads();
  for (int off = NSUB / 2; off > 0; off >>= 1) {
    if (s < off) red[s] += red[s + off];
    __syncthreads();
  }
  if (s == 0) out[0] = red[0] * (1.0f / NSUB);
}

// ---------------------------------------------------------------------------
extern "C" void kernel_launch(void* const* d_in, const int* in_sizes, int n_in,
                              void* d_out, int out_size, void* d_ws, size_t ws_size,
                              hipStream_t stream) {
  const float* x       = (const float*)d_in[0];
  const int*   esrc    = (const int*)d_in[1];
  const int*   edst    = (const int*)d_in[2];
  const float* ew      = (const float*)d_in[3];
  const int*   batches = (const int*)d_in[4];
  // d_in[5] segment_ids: consecutive (m // 64) by construction -> not needed
  const float* labels  = (const float*)d_in[6];
  const float* W0      = (const float*)d_in[7];
  const float* b0      = (const float*)d_in[8];
  const float* W1      = (const float*)d_in[9];
  const float* b1      = (const float*)d_in[10];
  const float* W2      = (const float*)d_in[11];
  const float* b2      = (const float*)d_in[12];
  const float* eps     = (const float*)d_in[13];
  const float* Wr1     = (const float*)d_in[14];
  const float* br1     = (const float*)d_in[15];
  const float* Wr2     = (const float*)d_in[16];
  const float* br2     = (const float*)d_in[17];

  float* hA = (float*)d_ws;                       // [N, 256] current features
  float* hB = hA + (size_t)N_NODES * HIDD;        // [N, 256] msg / U buffer
  float* z  = hB + (size_t)N_NODES * HIDD;        // [512, 1024] pooled
  float* r1 = z  + (size_t)NSUB * 4 * HIDD;       // [512, 256]
  float* Wp = r1 + (size_t)NSUB * HIDD;           // [132, 256] padded W0

  const int nh = N_NODES * HIDD;
  const int T = 256;

  build_h0_kernel<<<(nh + T - 1) / T, T, 0, stream>>>(x, hA);
  set_batch_feat_kernel<<<(MBAT + T - 1) / T, T, 0, stream>>>(batches, hA);
  pad_w0_kernel<<<(K0PAD * HIDD + T - 1) / T, T, 0, stream>>>(W0, Wp);

  const float* bs[3] = {b0, b1, b2};
  for (int l = 0; l < 3; ++l) {
    zero_kernel<<<(nh + T - 1) / T, T, 0, stream>>>(hB, nh);
    const int ng = (l == 0) ? 33 : 64;            // float4 groups (132/256 cols)
    const int tot = N_EDGES * ng;
    scatter_edges_kernel<<<(tot + T - 1) / T, T, 0, stream>>>(
        esrc, edst, ew, hA, hB, ng, tot);
    combine_kernel<<<(nh + T - 1) / T, T, 0, stream>>>(hA, hB, eps, l, nh);
    if (l == 0) {
      wmma_gemm_relu_kernel<K0PAD><<<N_NODES / 8, T, 0, stream>>>(
          hB, HIDD, Wp, bs[0], hA, HIDD);
    } else {
      wmma_gemm_relu_kernel<HIDD><<<N_NODES / 8, T, 0, stream>>>(
          hB, HIDD, (l == 1) ? W1 : W2, bs[l], hA, HIDD);
    }
  }

  multipool_kernel<<<NSUB, HIDD, 0, stream>>>(hA, batches, z);

  // r1 = relu(z @ Wr1 + br1): [512,1024] x [1024,256]
  wmma_gemm_relu_kernel<4 * HIDD><<<NSUB / 8, T, 0, stream>>>(
      z, 4 * HIDD, Wr1, br1, r1, HIDD);

  loss_kernel<<<1, NSUB, 0, stream>>>(r1, Wr2, br2, labels, (float*)d_out);
}